// LIF_Spike_67422396612611
// MI455X (gfx1250) — compile-verified
//
#include <hip/hip_runtime.h>
#include <cstdint>

// LIF spike scan: out[t] = spike train of  mem = mem*DECAY*(1-spk) + x[t]
// x: [T=16, B=64, C=128, H=32, W=32] f32 -> out same shape.
// Bandwidth-bound (1.07 GB traffic, ~46us floor at 23.3 TB/s).
// Strategy: per lane, stage all T timestep elements into LDS with
// global_load_async_to_lds_b128 (deep MLP, ASYNCcnt-tracked), drain in order
// with s_wait_asynccnt, keep the recurrence in registers, stream results out
// with non-temporal b128 stores.

typedef float f32x4 __attribute__((ext_vector_type(4)));

constexpr int   T       = 16;
constexpr int   BLOCK   = 256;
constexpr float kThresh = 0.5f;
constexpr float kDecay  = 0.2f;

__global__ __launch_bounds__(BLOCK, 1) void lif_spike_scan_kernel(
    const float* __restrict__ x, float* __restrict__ out, int nvec)
{
    // T-deep staging buffer: 16 * 256 * 16B = 64 KB of the 320 KB WGP LDS.
    __shared__ f32x4 tile[T][BLOCK];

    const int tid  = threadIdx.x;
    const int gvec = blockIdx.x * BLOCK + tid;     // float4 index within one timestep
    if (gvec >= nvec) return;

    const uint32_t tstride = (uint32_t)nvec * 16u; // bytes between timesteps (32 MB)
    const uint32_t base    = (uint32_t)gvec * 16u; // this lane's byte offset in a timestep

    // ---- Issue all T async global->LDS copies up front (ASYNCcnt = T = 16).
    // GVS mode: 64-bit uniform base in SGPRs + 32-bit per-lane offset VGPR.
    // Each lane writes (and later reads) only its own LDS slot, and per-wave
    // async loads complete in order, so s_wait_asynccnt alone synchronizes.
    #pragma unroll
    for (int t = 0; t < T; ++t) {
        uint32_t voff = base + (uint32_t)t * tstride;            // < 2^31
        uint32_t lds  = (uint32_t)(uintptr_t)&tile[t][tid];      // low 32b of generic LDS ptr = LDS byte addr
        asm volatile("global_load_async_to_lds_b128 %0, %1, %2"
                     :: "v"(lds), "v"(voff), "s"(x)
                     : "memory");
    }

    f32x4 mem = {0.f, 0.f, 0.f, 0.f};
    f32x4 spk = {0.f, 0.f, 0.f, 0.f};

    #pragma unroll
    for (int t = 0; t < T; ++t) {
        // Async loads retire in order: waiting ASYNCcnt <= (T-1-t) guarantees
        // copies 0..t have landed in LDS.
        asm volatile("s_wait_asynccnt %0" :: "i"(T - 1 - t) : "memory");

        f32x4 xv = tile[t][tid];                  // ds_load_b128

        #pragma unroll
        for (int c = 0; c < 4; ++c) {
            // mem = mem*DECAY*(1-spk) + x  ->  fma(mem*(1-spk), DECAY, x)
            float m = __builtin_fmaf(mem[c] * (1.0f - spk[c]), kDecay, xv[c]);
            mem[c] = m;
            spk[c] = (m > kThresh) ? 1.0f : 0.0f;
        }

        // Write-once 512 MB output stream: non-temporal so it doesn't evict
        // the read stream from L2.
        f32x4* dst = (f32x4*)out + (size_t)t * (size_t)nvec + (size_t)gvec;
        __builtin_nontemporal_store(spk, dst);
    }
}

extern "C" void kernel_launch(void* const* d_in, const int* in_sizes, int n_in,
                              void* d_out, int out_size, void* d_ws, size_t ws_size,
                              hipStream_t stream)
{
    const float* x   = (const float*)d_in[0];
    float*       out = (float*)d_out;

    const long long total = (long long)in_sizes[0];  // T*B*C*H*W = 134,217,728
    const int nvec   = (int)(total / T / 4);         // float4 slots per timestep = 2,097,152
    const int blocks = (nvec + BLOCK - 1) / BLOCK;   // 8192

    lif_spike_scan_kernel<<<blocks, BLOCK, 0, stream>>>(x, out, nvec);
}